// BlurModel_59502476919288
// MI455X (gfx1250) — compile-verified
//
#include <hip/hip_runtime.h>
#include <stdint.h>

// ---------------- constants ----------------
#define HW      3072
#define PATCH   384
#define NPIX    (PATCH * PATCH)      // 147456 per patch
#define NBINS   20480                // padded; grid step 5e-5 -> 20000 real bins
#define HIST_BYTES   (NBINS * 4)     // 81920 B
#define STAGE_FLOATS 4096            // floats per async chunk (1024 thr x 4)
#define STAGE_BYTES  (2 * STAGE_FLOATS * 4)   // 32768 B double buffer
#define NCHUNK  (NPIX / STAGE_FLOATS)         // 36 exact
#define STEPF   5e-5f

// CDNA5 async tensor path: per-lane 16B global->LDS DMA (GVS mode), ASYNCcnt.
__device__ __forceinline__ void async_load_b128_to_lds(uint32_t lds_off,
                                                       uint32_t goff,
                                                       const float* base) {
    asm volatile("global_load_async_to_lds_b128 %0, %1, %2"
                 :
                 : "v"(lds_off), "v"(goff), "s"(base)
                 : "memory");
}
__device__ __forceinline__ void wait_async_le1() {
    asm volatile("s_wait_asynccnt 0x1" ::: "memory");
}
__device__ __forceinline__ void wait_async_le0() {
    asm volatile("s_wait_asynccnt 0x0" ::: "memory");
}

// =====================================================================
// Kernel 1: per-patch threshold via LDS histogram + block suffix scan.
// One block per patch. Input streamed via async global->LDS B128 DMA,
// double-buffered; each wave only touches its own 512B slice, so the
// pipeline needs no cross-wave barriers (ASYNCcnt is per-wave).
// =====================================================================
extern "C" __global__ void __launch_bounds__(1024, 1)
cdna5_patch_threshold(const float* __restrict__ x, float* __restrict__ ths) {
    extern __shared__ uint8_t smem[];
    uint32_t* hist    = (uint32_t*)smem;                               // 81920 B
    float*    stage   = (float*)(smem + HIST_BYTES);                   // 32768 B
    uint32_t* partial = (uint32_t*)(smem + HIST_BYTES + STAGE_BYTES);  // 4096 B
    int*      minK    = (int*)(smem + HIST_BYTES + STAGE_BYTES + 4096);

    const int tid  = threadIdx.x;
    const int p    = blockIdx.x;
    const int pr   = p >> 3, pc = p & 7;
    const int row0 = pr * PATCH, col0 = pc * PATCH;

    for (int i = tid; i < NBINS; i += 1024) hist[i] = 0u;
    if (tid == 0) *minK = 20000;  // k=20000 (th=1.0) always satisfies
    __syncthreads();

    // LDS byte offset of staging buffer = low 32 bits of generic pointer
    // (flat LDS aperture keeps the group-relative offset in addr[31:0]).
    const uint32_t stage_lds = (uint32_t)(uintptr_t)(void*)stage;

    // ---- issue chunk 0 ----
    {
        int i = tid * 4;
        int r = i / PATCH, cc = i - r * PATCH;
        uint32_t goff = (uint32_t)(((row0 + r) * HW + col0 + cc) * 4);
        async_load_b128_to_lds(stage_lds + (uint32_t)(tid * 16), goff, x);
    }
    // ---- pipelined histogram ----
    for (int c = 0; c < NCHUNK; ++c) {
        if (c + 1 < NCHUNK) {
            int i = (c + 1) * STAGE_FLOATS + tid * 4;
            int r = i / PATCH, cc = i - r * PATCH;
            uint32_t goff = (uint32_t)(((row0 + r) * HW + col0 + cc) * 4);
            uint32_t loff = stage_lds +
                            (uint32_t)(((c + 1) & 1) * (STAGE_FLOATS * 4) + tid * 16);
            async_load_b128_to_lds(loff, goff, x);
            wait_async_le1();   // chunk c landed; chunk c+1 may be in flight
        } else {
            wait_async_le0();
        }
        const float* st = stage + (c & 1) * STAGE_FLOATS + tid * 4;
#pragma unroll
        for (int j = 0; j < 4; ++j) {
            float v = st[j];
            int b = (int)(v * 20000.0f);           // bin = floor(x / 5e-5)
            b = b < 0 ? 0 : (b > NBINS - 1 ? NBINS - 1 : b);
            atomicAdd(&hist[b], 1u);
        }
    }
    __syncthreads();

    // ---- suffix scan: cnt(k) = #pixels with bin >= k ----
    const int base = tid * 20;                     // 1024 x 20 = 20480 bins
    uint32_t s = 0;
#pragma unroll
    for (int j = 0; j < 20; ++j) s += hist[base + j];
    partial[tid] = s;
    __syncthreads();
    for (int off = 1; off < 1024; off <<= 1) {     // Hillis-Steele suffix scan
        uint32_t v = partial[tid];
        if (tid + off < 1024) v += partial[tid + off];
        __syncthreads();
        partial[tid] = v;
        __syncthreads();
    }

    const bool  frame = (pr == 0) | (pr == 7) | (pc == 0) | (pc == 7);
    const float hi    = 0.08f - (frame ? 0.05f : 0.0f);   // hi_tgt, f32 as ref
    uint32_t run = (tid < 1023) ? partial[tid + 1] : 0u;
    int best = 0x7FFFFFFF;
    for (int j = 19; j >= 0; --j) {                // descending k within chunk
        run += hist[base + j];
        int k = base + j;
        if (k < 20000 && ((float)run / 147456.0f <= hi)) best = k;
    }
    if (best != 0x7FFFFFFF) atomicMin(minK, best);
    __syncthreads();
    if (tid == 0) ths[p] = (float)(*minK) * STEPF;
}

// =====================================================================
// Kernel 2: fused binarize + morphological close (5x5 dilate, 5x5 erode)
// on bit-packed rows. 64-row x 1024-col tile per block, 8 waves.
// wave32 ballot packs 32 comparisons -> one bitmap word.
// =====================================================================
extern "C" __global__ void __launch_bounds__(256, 4)
cdna5_binarize_close(const float* __restrict__ x, const float* __restrict__ ths,
                     float* __restrict__ out) {
    __shared__ uint32_t A[72 * 36];   // bits, later D (dilated)
    __shared__ uint32_t B[72 * 34];   // H (hor OR5), later Eh (hor AND5)
    __shared__ float    sth[64];

    const int tid  = threadIdx.x, lane = tid & 31, wv = tid >> 5;
    const int r0   = blockIdx.y * 64;   // output rows r0..r0+63
    const int w0   = blockIdx.x * 32;   // output words w0..w0+31 (32 px/word)

    if (tid < 64) sth[tid] = ths[tid];
    __syncthreads();

    // ---- phase 1: binarize with 4-row / 2-word halo (zeros outside) ----
    for (int t = wv; t < 72 * 36; t += 8) {          // one word per wave-iter
        int i  = t / 36, j = t - i * 36;
        int gr = r0 - 4 + i;                          // global row
        int gw = w0 - 2 + j;                          // global word
        bool pred = false;
        if (((unsigned)gr < 3072u) & ((unsigned)gw < 96u)) {  // wave-uniform
            int col   = (gw << 5) + lane;
            float th  = sth[(gr / 384) * 8 + (gw / 12)];      // 384 % 32 == 0
            pred = x[gr * 3072 + col] > th;
        }
        uint32_t m = __builtin_amdgcn_ballot_w32(pred);
        if (lane == 0) A[t] = m;
    }
    __syncthreads();

    // ---- phase 2: horizontal OR over +-2 bits (dilate X), rows 72 x 34 ----
    for (int t = tid; t < 72 * 34; t += 256) {
        int i = t / 34, j2 = t - i * 34;
        const uint32_t* a = &A[i * 36 + j2];
        uint32_t l = a[0], m = a[1], r = a[2];
        B[t] = m | (m << 1) | (l >> 31) | (m << 2) | (l >> 30)
                 | (m >> 1) | (r << 31) | (m >> 2) | (r << 30);
    }
    __syncthreads();

    // ---- phase 3: vertical OR over 5 rows (dilate Y), rows 68 x 34 ----
    // Out-of-image rows have H == 0 (zeros in A), equal to clipped max.
    for (int t = tid; t < 68 * 34; t += 256) {
        int i3 = t / 34, j2 = t - i3 * 34;
        const uint32_t* b = &B[i3 * 34 + j2];
        A[i3 * 36 + j2] = b[0] | b[34] | b[68] | b[102] | b[136];
    }
    __syncthreads();

    // ---- phase 4: horizontal AND over +-2 bits (erode X), ones at edges ----
    for (int t = tid; t < 68 * 32; t += 256) {
        int i3 = t >> 5, j4 = t & 31;
        int gw = w0 + j4;
        const uint32_t* a = &A[i3 * 36 + j4];
        uint32_t l = (gw == 0)  ? 0xFFFFFFFFu : a[0];
        uint32_t m = a[1];
        uint32_t r = (gw == 95) ? 0xFFFFFFFFu : a[2];
        B[i3 * 34 + j4] = m & ((m << 1) | (l >> 31)) & ((m << 2) | (l >> 30))
                            & ((m >> 1) | (r << 31)) & ((m >> 2) | (r << 30));
    }
    __syncthreads();

    // ---- phase 5: vertical AND (erode Y, clipped) + expand bits -> f32 ----
    for (int t = wv; t < 64 * 32; t += 8) {
        int io = t >> 5, j4 = t & 31;
        int gr = r0 + io;
        uint32_t o = 0xFFFFFFFFu;
#pragma unroll
        for (int dr = -2; dr <= 2; ++dr) {
            int g = gr + dr;                          // wave-uniform clip
            if (g >= 0 && g < 3072) o &= B[(io + 2 + dr) * 34 + j4];
        }
        int col = ((w0 + j4) << 5) + lane;
        out[gr * 3072 + col] = ((o >> lane) & 1u) ? 1.0f : 0.0f;
    }
}

// =====================================================================
extern "C" void kernel_launch(void* const* d_in, const int* in_sizes, int n_in,
                              void* d_out, int out_size, void* d_ws, size_t ws_size,
                              hipStream_t stream) {
    (void)in_sizes; (void)n_in; (void)out_size; (void)ws_size;
    const float* x   = (const float*)d_in[0];
    float*       out = (float*)d_out;
    float*       ths = (float*)d_ws;   // 64 per-patch thresholds

    const size_t shmem1 = HIST_BYTES + STAGE_BYTES + 4096 + 16;  // ~118.8 KB
    cdna5_patch_threshold<<<dim3(64), dim3(1024), shmem1, stream>>>(x, ths);

    dim3 g2(HW / 1024, HW / 64);       // 3 x 48 tiles
    cdna5_binarize_close<<<g2, dim3(256), 0, stream>>>(x, ths, out);
}